// GreedyMentionProposer_34866544509522
// MI455X (gfx1250) — compile-verified
//
#include <hip/hip_runtime.h>
#include <hip/hip_bf16.h>
#include <stdint.h>

// ---------------------------------------------------------------------------
// GreedyMentionProposer for MI455X (gfx1250)
//   - wave32 ballot crossing test (span width <= 30 < 32 lanes)
//   - 320KB LDS: full sorted candidate list + interval tables resident
//   - TDM strided DMA (tensor_load_to_lds) to compact sorted ids into LDS
// ---------------------------------------------------------------------------

typedef unsigned int u32x4 __attribute__((ext_vector_type(4)));
typedef int          i32x4 __attribute__((ext_vector_type(4)));
typedef int          i32x8 __attribute__((ext_vector_type(8)));

constexpr int SPAN_T = 2048;                 // token_num
constexpr int SPAN_W = 30;                   // max span width
constexpr int SPAN_N = SPAN_T * SPAN_W;      // 61440 candidates
constexpr int SPAN_K = (SPAN_T * 4) / 10;    // 819 = int(0.4*T)
constexpr int NPAD   = 65536;                // next pow2 >= SPAN_N

// float -> unsigned whose ASCENDING order equals DESCENDING float order
__device__ __forceinline__ unsigned sortable_desc(float f) {
  unsigned u = __float_as_uint(f);
  u ^= ((unsigned)((int)u >> 31)) | 0x80000000u;  // ascending-sortable
  return ~u;                                      // invert -> descending
}

// ---- Phase 1: build 64-bit keys [ ~sortable(score+log(mask)) : cand_id ] ----
__global__ void init_keys(const float* __restrict__ scores,
                          const float* __restrict__ mask,
                          unsigned long long* __restrict__ keys) {
  int i = blockIdx.x * blockDim.x + threadIdx.x;
  if (i >= NPAD) return;
  if (i < SPAN_N) {
    float s = scores[i] + __logf(mask[i]);   // mask==1 -> +0, mask==0 -> -inf
    keys[i] = ((unsigned long long)sortable_desc(s) << 32) | (unsigned)i;
  } else {
    keys[i] = ~0ULL;                          // pad: sorts to the end
  }
}

// ---- Phase 2: bitonic sort (ascending) of 65536 keys ----------------------
// Fused in-LDS kernel: all phases k = 2..2048 (strides within a 2048 tile).
__global__ void bitonic_local_full(unsigned long long* __restrict__ keys) {
  __shared__ unsigned long long tile[2048];
  const int t = threadIdx.x;                  // 1024 threads / 2048 elements
  const int base = blockIdx.x * 2048;
  tile[t]        = keys[base + t];
  tile[t + 1024] = keys[base + t + 1024];
  __syncthreads();
  for (int k = 2; k <= 2048; k <<= 1) {
    for (int j = k >> 1; j >= 1; j >>= 1) {
      int i = ((t & ~(j - 1)) << 1) | (t & (j - 1));
      int p = i | j;
      bool up = (((base + i) & k) == 0);
      unsigned long long a = tile[i], b = tile[p];
      if ((a > b) == up) { tile[i] = b; tile[p] = a; }
      __syncthreads();
    }
  }
  keys[base + t]        = tile[t];
  keys[base + t + 1024] = tile[t + 1024];
}

// One global compare-swap step (stride j >= 2048, crosses LDS tiles).
__global__ void bitonic_global(unsigned long long* __restrict__ keys,
                               int j, int k) {
  int i = blockIdx.x * blockDim.x + threadIdx.x;
  int p = i ^ j;
  if (p > i) {
    bool up = ((i & k) == 0);
    unsigned long long a = keys[i], b = keys[p];
    if ((a > b) == up) { keys[i] = b; keys[p] = a; }
  }
}

// Fused in-LDS tail of phase k: strides 1024 .. 1 (direction uniform per tile).
__global__ void bitonic_local_merge(unsigned long long* __restrict__ keys,
                                    int k) {
  __shared__ unsigned long long tile[2048];
  const int t = threadIdx.x;
  const int base = blockIdx.x * 2048;
  tile[t]        = keys[base + t];
  tile[t + 1024] = keys[base + t + 1024];
  __syncthreads();
  const bool up = ((base & k) == 0);
  for (int j = 1024; j >= 1; j >>= 1) {
    int i = ((t & ~(j - 1)) << 1) | (t & (j - 1));
    int p = i | j;
    unsigned long long a = tile[i], b = tile[p];
    if ((a > b) == up) { tile[i] = b; tile[p] = a; }
    __syncthreads();
  }
  keys[base + t]        = tile[t];
  keys[base + t + 1024] = tile[t + 1024];
}

// ---- Phase 3: single-wave greedy non-crossing scan ------------------------
// LDS layout (dynamic, 262144 B): [0)   s2e[2048]   start_to_max_end
//                                 [8K)  e2s[2048]   end_to_min_start
//                                 [16K) idx[61440]  sorted candidate ids (TDM)
__global__ void greedy_scan(const unsigned long long* __restrict__ keys,
                            int* __restrict__ acc, int* __restrict__ cntp) {
  extern __shared__ int smem[];
  int* s2e = smem;          // [SPAN_T]
  int* e2s = smem + 2048;   // [SPAN_T]
  int* idx = smem + 4096;   // [SPAN_N], LDS byte offset 16384
  const int lane = threadIdx.x;     // 32 threads = 1 wave

  // --- TDM: strided gather of low dword of each sorted key -> idx[] in LDS.
  // D# group0: count=1, lds_addr=16384, global_addr=keys, type=2.
  // D# group1: data_size=4B, tensor_dim0=2, tensor_dim1=61440,
  //            tile_dim0=1, tile_dim1=61440, dim0_stride=2 elements
  //            -> compacts keys[i].lo32 into 61440 contiguous LDS dwords.
  unsigned long long ga = (unsigned long long)(uintptr_t)keys;
  u32x4 g0 = { 1u,                                     // [1:0] count=1
               16384u,                                 // [63:32] lds_addr
               (unsigned)(ga & 0xFFFFFFFFull),         // [95:64] gaddr lo
               (unsigned)((ga >> 32) & 0x01FFFFFFull)  // [120:96] gaddr hi
                 | 0x80000000u };                      // [127:126] type=2
  i32x8 g1 = { 0x00020000,        // [17:16] data_size=2 (4-byte)
               0x00020000,        // [63:48] tensor_dim0 = 2
               (int)0xF0000000u,  // [95:80] tensor_dim1 = 61440 (lo16)
               0x00010000,        // [127:112] tile_dim0 = 1
               0x0000F000,        // [143:128] tile_dim1 = 61440
               2,                 // [191:160] tensor_dim0_stride = 2
               0, 0 };
  i32x4 z4 = { 0, 0, 0, 0 };
#if __clang_major__ >= 23
  i32x8 z8 = { 0, 0, 0, 0, 0, 0, 0, 0 };
  __builtin_amdgcn_tensor_load_to_lds(g0, g1, z4, z4, z8, 0);
#else
  __builtin_amdgcn_tensor_load_to_lds(g0, g1, z4, z4, 0);
#endif

  // Init interval tables while the DMA is in flight.
  for (int t = lane; t < SPAN_T; t += 32) { s2e[t] = -1; e2s[t] = -1; }
  __builtin_amdgcn_s_wait_tensorcnt(0);
  __syncthreads();

  int cnt = 0;
  for (int pos = 0; pos < SPAN_N; ++pos) {
    int c = idx[pos];                   // LDS broadcast (all lanes same addr)
    int s = c / SPAN_W;
    int e = s + (c - s * SPAN_W);
    if (e > SPAN_T - 1) e = SPAN_T - 1;
    int width = e - s;                  // <= 29 < 32 lanes

    // wave32 one-shot range query: lane L probes token s+L
    int p = 0;
    if (lane < width) {
      int a = s2e[s + 1 + lane];        // tok in (s, e]
      int b = e2s[s + lane];            // tok in [s, e)
      p = (int)(a > e) | ((int)(b >= 0) & (int)(b < s));
    }
    if (!__any(p)) {                    // no crossing -> accept
      if (lane == 0) {
        int m = s2e[s];  s2e[s] = (e > m) ? e : m;
        int cur = e2s[e]; e2s[e] = (cur < 0 || s < cur) ? s : cur;
        acc[cnt] = c;
      }
      ++cnt;
      if (cnt >= SPAN_K) break;         // uniform: no further state changes
    }
    __syncthreads();                    // order lane-0 LDS updates vs reads
  }
  if (lane == 0) *cntp = cnt;
}

// ---- Phase 4: rank accepted spans by (start*T+end, id), scatter outputs ---
__global__ void finalize(const int* __restrict__ acc,
                         const int* __restrict__ cntp,
                         const float* __restrict__ scores,
                         float* __restrict__ out) {
  __shared__ int sacc[1024];
  int cnt = *cntp;
  if (cnt > 1024) cnt = 1024;           // safety (cnt <= 819)
  for (int t = threadIdx.x; t < cnt; t += blockDim.x) sacc[t] = acc[t];

  float* oS  = out;                     // top_scores [K]
  float* oI  = out + SPAN_K;            // top_idx    [K]
  float* oSp = out + 2 * SPAN_K;        // top_spans  [K,2]
  float* oV  = out + 4 * SPAN_K;        // valid      [K]
  for (int t = threadIdx.x; t < SPAN_K; t += blockDim.x) {
    oS[t] = 0.f; oI[t] = 0.f; oV[t] = 0.f;
    oSp[2 * t] = 0.f; oSp[2 * t + 1] = 0.f;
  }
  __syncthreads();

  for (int t = threadIdx.x; t < cnt; t += blockDim.x) {
    int c = sacc[t];
    int s = c / SPAN_W;
    int e = s + (c - s * SPAN_W);
    if (e > SPAN_T - 1) e = SPAN_T - 1;
    int key = s * SPAN_T + e;
    int rank = 0;
    for (int b = 0; b < cnt; ++b) {
      int cb = sacc[b];
      int sb = cb / SPAN_W;
      int eb = sb + (cb - sb * SPAN_W);
      if (eb > SPAN_T - 1) eb = SPAN_T - 1;
      int kb = sb * SPAN_T + eb;
      rank += (int)((kb < key) || (kb == key && cb < c));  // stable argsort
    }
    oS[rank] = scores[c];
    oI[rank] = (float)c;
    oSp[2 * rank]     = (float)s;
    oSp[2 * rank + 1] = (float)e;
    oV[rank] = 1.0f;
  }
}

// ---------------------------------------------------------------------------
extern "C" void kernel_launch(void* const* d_in, const int* in_sizes, int n_in,
                              void* d_out, int out_size, void* d_ws,
                              size_t ws_size, hipStream_t stream) {
  const float* scores = (const float*)d_in[1];   // span_mention_scores
  const float* mask   = (const float*)d_in[2];   // span_mask

  unsigned long long* keys = (unsigned long long*)d_ws;          // 512 KB
  int* acc  = (int*)((char*)d_ws + (size_t)NPAD * 8);            // 4 KB
  int* cntp = (int*)((char*)d_ws + (size_t)NPAD * 8 + 4096);

  init_keys<<<NPAD / 1024, 1024, 0, stream>>>(scores, mask, keys);
  bitonic_local_full<<<NPAD / 2048, 1024, 0, stream>>>(keys);
  for (int k = 4096; k <= NPAD; k <<= 1) {
    for (int j = k >> 1; j >= 2048; j >>= 1)
      bitonic_global<<<NPAD / 1024, 1024, 0, stream>>>(keys, j, k);
    bitonic_local_merge<<<NPAD / 2048, 1024, 0, stream>>>(keys, k);
  }
  // 16 KB tables + 240 KB sorted ids = 256 KB LDS (<= 320 KB / WGP on CDNA5)
  size_t smem = 16384 + (size_t)SPAN_N * 4;
  greedy_scan<<<1, 32, smem, stream>>>(keys, acc, cntp);
  finalize<<<1, 1024, 0, stream>>>(acc, cntp, scores, (float*)d_out);
}